// GIN2_48361331753437
// MI455X (gfx1250) — compile-verified
//
#include <hip/hip_runtime.h>
#include <hip/hip_bf16.h>

// ---------------------------------------------------------------------------
// GIN forward for MI455X (gfx1250, wave32).
// f32 WMMA (V_WMMA_F32_16X16X4_F32) for all dense GEMMs, f32 global atomics
// for the edge aggregation (L2-resident), LDS staging between fused GEMMs.
// gin_layer uses 80-row blocks: each wave reuses one B-fragment across 5 row
// tiles (5 WMMAs per weight fetch) to amortize weight traffic 5x.
// ---------------------------------------------------------------------------

#define NNODES 50000
#define NEDGES 800000
#define DIM    64
#define DIM2   32
#define NC     6
#define ROWT   5          // row tiles per block in gin_layer (80 rows)

typedef __attribute__((ext_vector_type(2))) float v2f;
typedef __attribute__((ext_vector_type(8))) float v8f;

// ---------------------------------------------------------------------------
// agg = h   (self term of GIN; eps == 0)
// ---------------------------------------------------------------------------
__global__ __launch_bounds__(256) void init_agg_kernel(const float* __restrict__ h,
                                                       float* __restrict__ agg) {
    int i = blockIdx.x * 256 + threadIdx.x;           // over N*64/4 float4s
    reinterpret_cast<float4*>(agg)[i] =
        reinterpret_cast<const float4*>(h)[i];
}

// ---------------------------------------------------------------------------
// agg[dst] += h[src] over all edges. One thread per (edge, 4-feature chunk).
// E*16 threads; float4 gather + 4x global_atomic_add_f32 (stays in L2).
// ---------------------------------------------------------------------------
__global__ __launch_bounds__(256) void scatter_kernel(const float* __restrict__ h,
                                                      const int* __restrict__ src,
                                                      const int* __restrict__ dst,
                                                      float* __restrict__ agg) {
    int tid = blockIdx.x * 256 + threadIdx.x;         // E*16 threads total
    int e = tid >> 4;
    int c = (tid & 15) << 2;
    int s = src[e];
    int d = dst[e];
    float4 v = *reinterpret_cast<const float4*>(h + (size_t)s * DIM + c);
    float* a = agg + (size_t)d * DIM + c;
    atomicAdd(a + 0, v.x);
    atomicAdd(a + 1, v.y);
    atomicAdd(a + 2, v.z);
    atomicAdd(a + 3, v.w);
}

// ---------------------------------------------------------------------------
// Fused GIN MLP: hout = relu( relu(BN(agg@Wa + ba)) @ Wb + bb )
// Block = 128 threads (4 waves) handles an 80-node row tile (5 x 16 rows).
// Wave w computes output columns [16w, 16w+16); one B-fragment per k-step is
// reused for 5 chained WMMAs (one per row tile). GEMM1 result staged in LDS.
// ---------------------------------------------------------------------------
__global__ __launch_bounds__(128) void gin_layer_kernel(
        const float* __restrict__ agg,   // [N,64]
        const float* __restrict__ Wa,    // [64,64] row-major (k,n)
        const float* __restrict__ ba,    // [64]
        const float* __restrict__ gbn,   // [64]
        const float* __restrict__ bebn,  // [64]
        const float* __restrict__ Wb,    // [64,64]
        const float* __restrict__ bb,    // [64]
        float* __restrict__ hout)        // [N,64]
{
    __shared__ float zs[16 * ROWT][DIM + 4];   // GEMM1 output tiles (padded)

    const int row0 = blockIdx.x * (16 * ROWT);
    const int lane = threadIdx.x & 31;
    const int wave = threadIdx.x >> 5;
    const int col0 = wave * 16;
    const int lm   = lane & 15;          // row (A) / col (B,C)
    const int lk   = (lane >> 4) << 1;   // K sub-offset {0,2}
    const int mtop = (lane >> 4) * 8;    // C/D: m = r + 8*(lane>>4)

    __builtin_prefetch(Wa, 0, 3);        // global_prefetch_b8
    __builtin_prefetch(Wb, 0, 3);

    // ---- GEMM1: Z = AGG @ Wa --------------------------------------------
    v8f acc[ROWT];
#pragma unroll
    for (int t = 0; t < ROWT; ++t) acc[t] = v8f{0.f, 0.f, 0.f, 0.f, 0.f, 0.f, 0.f, 0.f};

    const float* arow = agg + (size_t)(row0 + lm) * DIM;  // row tile 0 base
#pragma unroll
    for (int k0 = 0; k0 < DIM; k0 += 4) {
        v2f bf;                                          // shared across tiles
        bf.x = Wa[(k0 + lk) * DIM + col0 + lm];
        bf.y = Wa[(k0 + lk + 1) * DIM + col0 + lm];
#pragma unroll
        for (int t = 0; t < ROWT; ++t) {
            v2f af = *reinterpret_cast<const v2f*>(arow + (size_t)t * 16 * DIM + k0 + lk);
            acc[t] = __builtin_amdgcn_wmma_f32_16x16x4_f32(
                /*neg_a=*/false, af, /*neg_b=*/false, bf,
                /*c_mod=*/(short)0, acc[t], /*reuse_a=*/false, /*reuse_b=*/false);
        }
    }

    // ---- epilogue 1: +bias, eval-mode BN, ReLU -> LDS --------------------
    {
        const int col     = col0 + lm;
        const float bias  = ba[col];
        const float scale = gbn[col] * rsqrtf(1.0f + 1e-5f);
        const float shift = bebn[col];
#pragma unroll
        for (int t = 0; t < ROWT; ++t)
#pragma unroll
            for (int r = 0; r < 8; ++r) {
                float v = (acc[t][r] + bias) * scale + shift;
                zs[t * 16 + mtop + r][col] = v > 0.f ? v : 0.f;
            }
    }
    __syncthreads();

    // ---- GEMM2: H = relu(Z @ Wb + bb) ------------------------------------
    v8f acc2[ROWT];
#pragma unroll
    for (int t = 0; t < ROWT; ++t) acc2[t] = v8f{0.f, 0.f, 0.f, 0.f, 0.f, 0.f, 0.f, 0.f};

#pragma unroll
    for (int k0 = 0; k0 < DIM; k0 += 4) {
        v2f bf;
        bf.x = Wb[(k0 + lk) * DIM + col0 + lm];
        bf.y = Wb[(k0 + lk + 1) * DIM + col0 + lm];
#pragma unroll
        for (int t = 0; t < ROWT; ++t) {
            v2f af;
            af.x = zs[t * 16 + lm][k0 + lk];
            af.y = zs[t * 16 + lm][k0 + lk + 1];
            acc2[t] = __builtin_amdgcn_wmma_f32_16x16x4_f32(
                false, af, false, bf, (short)0, acc2[t], false, false);
        }
    }
    {
        const int col    = col0 + lm;
        const float bias = bb[col];
#pragma unroll
        for (int t = 0; t < ROWT; ++t)
#pragma unroll
            for (int r = 0; r < 8; ++r) {
                float v = acc2[t][r] + bias;
                hout[(size_t)(row0 + t * 16 + mtop + r) * DIM + col] = v > 0.f ? v : 0.f;
            }
    }
}

// ---------------------------------------------------------------------------
// Head: h3 = relu(h @ Wl1 + bl1)   [N,64] @ [64,32] -> [N,32]
// Block = 64 threads (2 waves) per 16-node tile; wave w -> cols [16w,16w+16).
// ---------------------------------------------------------------------------
__global__ __launch_bounds__(64) void head_kernel(const float* __restrict__ h,
                                                  const float* __restrict__ Wl1,
                                                  const float* __restrict__ bl1,
                                                  float* __restrict__ h3) {
    const int row0 = blockIdx.x * 16;
    const int lane = threadIdx.x & 31;
    const int wave = threadIdx.x >> 5;
    const int col0 = wave * 16;
    const int lm   = lane & 15;
    const int lk   = (lane >> 4) << 1;
    const int mtop = (lane >> 4) * 8;

    v8f acc = {0.f, 0.f, 0.f, 0.f, 0.f, 0.f, 0.f, 0.f};
    const float* arow = h + (size_t)(row0 + lm) * DIM;
#pragma unroll
    for (int k0 = 0; k0 < DIM; k0 += 4) {
        v2f af, bf;
        v2f ld = *reinterpret_cast<const v2f*>(arow + k0 + lk);
        af.x = ld.x;
        af.y = ld.y;
        bf.x = Wl1[(k0 + lk) * DIM2 + col0 + lm];
        bf.y = Wl1[(k0 + lk + 1) * DIM2 + col0 + lm];
        acc = __builtin_amdgcn_wmma_f32_16x16x4_f32(
            false, af, false, bf, (short)0, acc, false, false);
    }
    const int col    = col0 + lm;
    const float bias = bl1[col];
#pragma unroll
    for (int r = 0; r < 8; ++r) {
        float v = acc[r] + bias;
        h3[(size_t)(row0 + mtop + r) * DIM2 + col] = v > 0.f ? v : 0.f;
    }
}

// ---------------------------------------------------------------------------
// Edge head: e = [h3[src] || h3[dst]]  (written out) ; out = e @ Wl2 + bl2.
// One thread per edge; Wl2/bl2 staged in LDS; float4 gathers/stores.
// ---------------------------------------------------------------------------
__global__ __launch_bounds__(256) void edge_kernel(const float* __restrict__ h3,
                                                   const int* __restrict__ src,
                                                   const int* __restrict__ dst,
                                                   const float* __restrict__ Wl2,
                                                   const float* __restrict__ bl2,
                                                   float* __restrict__ out,    // [E,6]
                                                   float* __restrict__ eout) { // [E,64]
    __shared__ float sW[DIM * NC];
    __shared__ float sb[NC];
    for (int i = threadIdx.x; i < DIM * NC; i += 256) sW[i] = Wl2[i];
    if (threadIdx.x < NC) sb[threadIdx.x] = bl2[threadIdx.x];
    __syncthreads();

    const int e = blockIdx.x * 256 + threadIdx.x;     // E == 3125*256 exactly
    const int s = src[e];
    const int d = dst[e];

    float o[NC];
#pragma unroll
    for (int c = 0; c < NC; ++c) o[c] = sb[c];

    float4* erow = reinterpret_cast<float4*>(eout + (size_t)e * (2 * DIM2));
    const float4* hs = reinterpret_cast<const float4*>(h3 + (size_t)s * DIM2);
    const float4* hd = reinterpret_cast<const float4*>(h3 + (size_t)d * DIM2);

#pragma unroll
    for (int j = 0; j < 8; ++j) {
        float4 v = hs[j];
        erow[j] = v;
        const int k = j * 4;
#pragma unroll
        for (int c = 0; c < NC; ++c)
            o[c] += v.x * sW[(k + 0) * NC + c] + v.y * sW[(k + 1) * NC + c] +
                    v.z * sW[(k + 2) * NC + c] + v.w * sW[(k + 3) * NC + c];
    }
#pragma unroll
    for (int j = 0; j < 8; ++j) {
        float4 v = hd[j];
        erow[8 + j] = v;
        const int k = DIM2 + j * 4;
#pragma unroll
        for (int c = 0; c < NC; ++c)
            o[c] += v.x * sW[(k + 0) * NC + c] + v.y * sW[(k + 1) * NC + c] +
                    v.z * sW[(k + 2) * NC + c] + v.w * sW[(k + 3) * NC + c];
    }
#pragma unroll
    for (int c = 0; c < NC; ++c) out[(size_t)e * NC + c] = o[c];
}

// ---------------------------------------------------------------------------
// Host-side launch sequence (graph-capture safe: kernels only, all on stream)
// ---------------------------------------------------------------------------
extern "C" void kernel_launch(void* const* d_in, const int* in_sizes, int n_in,
                              void* d_out, int out_size, void* d_ws, size_t ws_size,
                              hipStream_t stream) {
    (void)in_sizes; (void)n_in; (void)out_size; (void)ws_size;

    const float* x    = (const float*)d_in[0];        // [N,64]
    const int*   eidx = (const int*)  d_in[1];        // [2,E]
    const int*   src  = eidx;
    const int*   dstp = eidx + NEDGES;
    // d_in[2] = batch (unused)
    const float* W1  = (const float*)d_in[3];
    const float* b1  = (const float*)d_in[4];
    const float* g1  = (const float*)d_in[5];
    const float* be1 = (const float*)d_in[6];
    const float* W2  = (const float*)d_in[7];
    const float* b2  = (const float*)d_in[8];
    const float* Wa  = (const float*)d_in[9];         // [3,64,64]
    const float* ba  = (const float*)d_in[10];        // [3,64]
    const float* ga  = (const float*)d_in[11];
    const float* bea = (const float*)d_in[12];
    const float* Wb  = (const float*)d_in[13];        // [3,64,64]
    const float* bb  = (const float*)d_in[14];
    const float* Wl1 = (const float*)d_in[15];        // [64,32]
    const float* bl1 = (const float*)d_in[16];
    const float* Wl2 = (const float*)d_in[17];        // [64,6]
    const float* bl2 = (const float*)d_in[18];

    // workspace layout
    const size_t N64 = (size_t)NNODES * DIM;
    float* hA  = (float*)d_ws;                        // [N,64] layer output
    float* agg = hA + N64;                            // [N,64] aggregation
    float* h3  = agg + N64;                           // [N,32] head output

    float* outp = (float*)d_out;                      // [E,6]
    float* eout = outp + (size_t)NEDGES * NC;         // [E,64]

    const int gInit    = (NNODES * DIM / 4) / 256;    // 3125
    const int gScatter = (NEDGES * 16) / 256;         // 50000
    const int gLayer   = NNODES / (16 * ROWT);        // 625
    const int gHead    = NNODES / 16;                 // 3125
    const int gEdges   = NEDGES / 256;                // 3125

    const float* hin = x;
    for (int l = 0; l < 4; ++l) {
        const float *wa, *biasa, *g, *be, *wb, *biasb;
        if (l == 0) { wa = W1; biasa = b1; g = g1; be = be1; wb = W2; biasb = b2; }
        else {
            const int i = l - 1;
            wa    = Wa  + (size_t)i * DIM * DIM;
            biasa = ba  + (size_t)i * DIM;
            g     = ga  + (size_t)i * DIM;
            be    = bea + (size_t)i * DIM;
            wb    = Wb  + (size_t)i * DIM * DIM;
            biasb = bb  + (size_t)i * DIM;
        }
        init_agg_kernel<<<gInit, 256, 0, stream>>>(hin, agg);
        scatter_kernel<<<gScatter, 256, 0, stream>>>(hin, src, dstp, agg);
        gin_layer_kernel<<<gLayer, 128, 0, stream>>>(agg, wa, biasa, g, be, wb, biasb, hA);
        hin = hA;
    }

    head_kernel<<<gHead, 64, 0, stream>>>(hA, Wl1, bl1, h3);
    edge_kernel<<<gEdges, 256, 0, stream>>>(h3, src, dstp, Wl2, bl2, outp, eout);
}